// FastGuidedFilter_23021024707311
// MI455X (gfx1250) — compile-verified
//
#include <hip/hip_runtime.h>
#include <stdint.h>

typedef float v4f __attribute__((ext_vector_type(4)));

#define R       8
#define LR      512
#define HR      2048
#define TILE    32
#define PATCH   (TILE + 2 * R)   // 48
#define PSTRIDE (PATCH + 4)      // 52: keeps every 4-col group 16B aligned
#define NPLANES 12               // 4 batch * 3 channel

__device__ __forceinline__ int imin(int a, int b) { return a < b ? a : b; }
__device__ __forceinline__ int imax(int a, int b) { return a > b ? a : b; }

// ---------------------------------------------------------------------------
// Kernel 1: per 32x32 tile, stage a 48x48 halo patch of lr_x / lr_y into LDS
// via async global->LDS DMA (b128 fast path for interior tiles, b32 + zero
// fill at image borders), then separable sliding-window 17x17 box sums of
// {x, y, x*y, x*x}, divide by analytic clamped-window counts, emit A and b.
// ---------------------------------------------------------------------------
__global__ __launch_bounds__(256) void gf_stats_kernel(
    const float* __restrict__ lr_x, const float* __restrict__ lr_y,
    float* __restrict__ A_out, float* __restrict__ B_out)
{
    __shared__ float sx[PATCH][PSTRIDE];
    __shared__ float sy[PATCH][PSTRIDE];
    __shared__ float hs[4][PATCH][TILE];       // horizontal box sums
    __shared__ float vs[4][TILE][TILE];        // full 2D box sums

    const int tid   = threadIdx.x;
    const int tx    = blockIdx.x;
    const int ty    = blockIdx.y;
    const int plane = blockIdx.z;
    const int gx0   = tx * TILE - R;           // patch origin in image coords
    const int gy0   = ty * TILE - R;

    const unsigned long long basex =
        (unsigned long long)(uintptr_t)(lr_x + (size_t)plane * LR * LR);
    const unsigned long long basey =
        (unsigned long long)(uintptr_t)(lr_y + (size_t)plane * LR * LR);

    const bool interior = (gx0 >= 0) && (gy0 >= 0) &&
                          (gx0 + PATCH <= LR) && (gy0 + PATCH <= LR);

    if (interior) {
        // ---- b128 async DMA: 12 loads per 48-float row, all 16B aligned ----
        for (int e = tid; e < PATCH * (PATCH / 4); e += 256) {   // 576 groups
            const int pr = e / (PATCH / 4);
            const int pc = (e % (PATCH / 4)) * 4;
            const uint32_t ldsx = (uint32_t)(uintptr_t)(void*)&sx[pr][pc];
            const uint32_t ldsy = (uint32_t)(uintptr_t)(void*)&sy[pr][pc];
            const int voff = ((gy0 + pr) * LR + (gx0 + pc)) * 4;
            asm volatile("global_load_async_to_lds_b128 %0, %1, %2 offset:0"
                         :: "v"(ldsx), "v"(voff), "s"(basex) : "memory");
            asm volatile("global_load_async_to_lds_b128 %0, %1, %2 offset:0"
                         :: "v"(ldsy), "v"(voff), "s"(basey) : "memory");
        }
    } else {
        // ---- border tiles: per-texel DMA, zero-fill outside the image ----
        for (int e = tid; e < PATCH * PATCH; e += 256) {   // 2304 = 9 * 256
            const int pr = e / PATCH, pc = e % PATCH;
            const int gr = gy0 + pr,  gc = gx0 + pc;
            if (gr >= 0 && gr < LR && gc >= 0 && gc < LR) {
                const uint32_t ldsx = (uint32_t)(uintptr_t)(void*)&sx[pr][pc];
                const uint32_t ldsy = (uint32_t)(uintptr_t)(void*)&sy[pr][pc];
                const int voff = (gr * LR + gc) * 4;
                asm volatile("global_load_async_to_lds_b32 %0, %1, %2 offset:0"
                             :: "v"(ldsx), "v"(voff), "s"(basex) : "memory");
                asm volatile("global_load_async_to_lds_b32 %0, %1, %2 offset:0"
                             :: "v"(ldsy), "v"(voff), "s"(basey) : "memory");
            } else {
                sx[pr][pc] = 0.0f;   // zero outside image == truncated window
                sy[pr][pc] = 0.0f;
            }
        }
    }
    asm volatile("s_wait_asynccnt 0x0" ::: "memory");
    __syncthreads();

    // ---- Horizontal sliding 17-tap sums: one thread per (row, quantity) ----
    if (tid < 4 * PATCH) {                 // 192 threads active
        const int row = tid % PATCH;
        const int q   = tid / PATCH;       // 0:x 1:y 2:x*y 3:x*x
        auto val = [&](int c) -> float {
            const float x = sx[row][c], y = sy[row][c];
            return (q == 0) ? x : (q == 1) ? y : (q == 2) ? x * y : x * x;
        };
        float s = 0.0f;
        #pragma unroll
        for (int t = 0; t <= 2 * R; ++t) s += val(t);
        hs[q][row][0] = s;
        for (int j = 1; j < TILE; ++j) {
            s += val(j + 2 * R) - val(j - 1);
            hs[q][row][j] = s;
        }
    }
    __syncthreads();

    // ---- Vertical sliding 17-tap sums: one thread per (col, quantity) ----
    if (tid < 4 * TILE) {                  // 128 threads active
        const int j = tid % TILE;
        const int q = tid / TILE;
        float s = 0.0f;
        #pragma unroll
        for (int t = 0; t <= 2 * R; ++t) s += hs[q][t][j];
        vs[q][0][j] = s;
        for (int i = 1; i < TILE; ++i) {
            s += hs[q][i + 2 * R][j] - hs[q][i - 1][j];
            vs[q][i][j] = s;
        }
    }
    __syncthreads();

    // ---- Per-pixel statistics -> A, b (analytic window counts) ----
    for (int e = tid; e < TILE * TILE; e += 256) {
        const int i  = e >> 5, j = e & (TILE - 1);
        const int gy = ty * TILE + i, gx = tx * TILE + j;
        const float cy   = (float)(imin(gy + R, LR - 1) - imax(gy - R, 0) + 1);
        const float cx   = (float)(imin(gx + R, LR - 1) - imax(gx - R, 0) + 1);
        const float invN = 1.0f / (cy * cx);
        const float mx   = vs[0][i][j] * invN;
        const float my   = vs[1][i][j] * invN;
        const float cov  = vs[2][i][j] * invN - mx * my;
        const float var  = vs[3][i][j] * invN - mx * mx;
        const float a    = cov / (var + 1e-8f);
        const float b    = my - a * mx;
        const size_t o = (size_t)plane * LR * LR + (size_t)gy * LR + gx;
        A_out[o] = a;      // RT stores: A/b stay L2-resident (25 MB << 192 MB)
        B_out[o] = b;
    }
}

// ---------------------------------------------------------------------------
// Kernel 2: bilinear (align_corners) upsample of A,b to 2048^2 fused with
// out = A*hr + b. hr_x / out are streamed with non-temporal hints so the
// 400 MB stream does not evict the L2-resident A/b planes.
// ---------------------------------------------------------------------------
__global__ __launch_bounds__(256) void gf_apply_kernel(
    const float* __restrict__ hr_x, const float* __restrict__ A,
    const float* __restrict__ B, float* __restrict__ out)
{
    const int plane = blockIdx.z;
    const int Y     = blockIdx.y;
    const int X0    = (blockIdx.x * 256 + threadIdx.x) * 4;

    const double fy = (double)Y * 511.0 / 2047.0;   // exact at endpoints
    int y0 = (int)fy;          if (y0 > LR - 1) y0 = LR - 1;
    int y1 = y0 + 1;           if (y1 > LR - 1) y1 = LR - 1;
    const float wy = (float)(fy - (double)y0);

    const float* Ap = A + (size_t)plane * LR * LR;
    const float* Bp = B + (size_t)plane * LR * LR;
    const float* A0 = Ap + (size_t)y0 * LR;
    const float* A1 = Ap + (size_t)y1 * LR;
    const float* B0 = Bp + (size_t)y0 * LR;
    const float* B1 = Bp + (size_t)y1 * LR;

    const size_t hro = (size_t)plane * HR * HR + (size_t)Y * HR + X0;
    const v4f h = __builtin_nontemporal_load((const v4f*)(hr_x + hro));
    v4f o;
    #pragma unroll
    for (int k = 0; k < 4; ++k) {
        const int X = X0 + k;
        const double fx = (double)X * 511.0 / 2047.0;
        int x0 = (int)fx;      if (x0 > LR - 1) x0 = LR - 1;
        int x1 = x0 + 1;       if (x1 > LR - 1) x1 = LR - 1;
        const float wx = (float)(fx - (double)x0);

        const float a00 = A0[x0], a01 = A0[x1], a10 = A1[x0], a11 = A1[x1];
        const float b00 = B0[x0], b01 = B0[x1], b10 = B1[x0], b11 = B1[x1];
        const float av = (a00 * (1.0f - wx) + a01 * wx) * (1.0f - wy)
                       + (a10 * (1.0f - wx) + a11 * wx) * wy;
        const float bv = (b00 * (1.0f - wx) + b01 * wx) * (1.0f - wy)
                       + (b10 * (1.0f - wx) + b11 * wx) * wy;
        o[k] = av * h[k] + bv;
    }
    __builtin_nontemporal_store(o, (v4f*)(out + hro));
}

// ---------------------------------------------------------------------------
extern "C" void kernel_launch(void* const* d_in, const int* in_sizes, int n_in,
                              void* d_out, int out_size, void* d_ws, size_t ws_size,
                              hipStream_t stream)
{
    const float* lr_x = (const float*)d_in[0];   // (4,3,512,512)
    const float* lr_y = (const float*)d_in[1];   // (4,3,512,512)
    const float* hr_x = (const float*)d_in[2];   // (4,3,2048,2048)
    float*       outp = (float*)d_out;           // (4,3,2048,2048)

    float* Aws = (float*)d_ws;                              // 12.6 MB
    float* Bws = Aws + (size_t)NPLANES * LR * LR;           // 12.6 MB

    gf_stats_kernel<<<dim3(LR / TILE, LR / TILE, NPLANES), 256, 0, stream>>>(
        lr_x, lr_y, Aws, Bws);

    gf_apply_kernel<<<dim3(HR / (256 * 4), HR, NPLANES), 256, 0, stream>>>(
        hr_x, Aws, Bws, outp);
}